// BestBuddyLoss_31413390802978
// MI455X (gfx1250) — compile-verified
//
#include <hip/hip_runtime.h>
#include <hip/hip_bf16.h>

typedef __attribute__((ext_vector_type(2))) float v2f;
typedef __attribute__((ext_vector_type(8))) float v8f;

#define B_      4
#define HIN     192
#define HO1     64
#define N1      (HO1*HO1)        // 4096 rows of p1/p2
#define MROWS   5376             // 4096 + 1024 + 256
#define JT      (MROWS/16)       // 336 j-tiles (even -> unroll-by-2 pipeline)
#define KPAD    32
#define DVAL    27
#define INV_CNT (1.0f/(4.0f*4096.0f*27.0f))

// ---------------- bicubic downsample (a=-0.75 taps at half-integer offsets) --
__global__ void bicubic_down(const float* __restrict__ src, float* __restrict__ dst,
                             int Hin, int Hout, int f, int start) {
    int t = blockIdx.x * blockDim.x + threadIdx.x;
    int total = 12 * Hout * Hout;               // B*C = 12 planes
    if (t >= total) return;
    int ox = t % Hout;
    int oy = (t / Hout) % Hout;
    int bc = t / (Hout * Hout);
    const float w[4] = {-0.09375f, 0.59375f, 0.59375f, -0.09375f};
    const float* s = src + (size_t)bc * Hin * Hin;
    float acc = 0.0f;
    #pragma unroll
    for (int ty = 0; ty < 4; ++ty) {
        int iy = f * oy + start + ty;
        iy = iy < 0 ? 0 : (iy > Hin - 1 ? Hin - 1 : iy);
        float r = 0.0f;
        #pragma unroll
        for (int tx = 0; tx < 4; ++tx) {
            int ix = f * ox + start + tx;
            ix = ix < 0 ? 0 : (ix > Hin - 1 ? Hin - 1 : ix);
            r += w[tx] * s[iy * Hin + ix];
        }
        acc += w[ty] * r;
    }
    dst[(size_t)bc * Hout * Hout + oy * Hout + ox] = acc;
}

// ---------------- unfold 3x3 stride-3 patches into [rows,32] + squared norms -
__global__ void unfold_pack(const float* __restrict__ src, int Hin, int Ho,
                            float* __restrict__ dst, float* __restrict__ nrm,
                            int ldRows, int rowOff) {
    int t = blockIdx.x * blockDim.x + threadIdx.x;
    int rows = Ho * Ho;
    if (t >= 4 * rows) return;
    int b = t / rows, r = t % rows;
    int oy = r / Ho, ox = r % Ho;
    float* drow = dst + ((size_t)b * ldRows + rowOff + r) * KPAD;
    float nn = 0.0f;
    int d = 0;
    #pragma unroll
    for (int c = 0; c < 3; ++c)
        #pragma unroll
        for (int kh = 0; kh < 3; ++kh)
            #pragma unroll
            for (int kw = 0; kw < 3; ++kw) {
                float v = src[(((size_t)b * 3 + c) * Hin + (oy * 3 + kh)) * Hin + (ox * 3 + kw)];
                drow[d++] = v;
                nn += v * v;
            }
    for (; d < KPAD; ++d) drow[d] = 0.0f;
    nrm[b * ldRows + rowOff + r] = nn;
}

// ---------------- fused WMMA GEMM + clamped-distance argmin --------------------
// one wave32 per (batch, 16-row i-tile); f32 16x16x4 WMMA over K=32 (8 steps),
// j-loop software-pipelined with register double-buffering of the B tiles.
__global__ __launch_bounds__(32)
void argmin_wmma(const float* __restrict__ P1, const float* __restrict__ Y,
                 const float* __restrict__ x1n, const float* __restrict__ y2n,
                 int* __restrict__ ind) {
    const int lane = threadIdx.x;        // 0..31
    const int h    = lane >> 4;          // half-wave
    const int m16  = lane & 15;
    const int b    = blockIdx.y;
    const int i0   = blockIdx.x * 16;

    // A-operand preload: lane holds A[m][2h], A[m][2h+1] per 4-wide K step
    const float* p1row = P1 + ((size_t)b * N1    + i0 + m16) * KPAD;
    const float* p2row = Y  + ((size_t)b * MROWS + i0 + m16) * KPAD;   // p2 = first block of Y
    v2f a1[8], a2[8];
    #pragma unroll
    for (int s = 0; s < 8; ++s) {
        int kb = 4 * s + 2 * h;
        a1[s] = *(const v2f*)(p1row + kb);
        a2[s] = *(const v2f*)(p2row + kb);
    }
    // per-lane norms of the 8 C rows this lane owns (m = g + 8h)
    float x1v[8], x2v[8];
    #pragma unroll
    for (int g = 0; g < 8; ++g) {
        int m = g + 8 * h;
        x1v[g] = x1n[b * N1    + i0 + m];
        x2v[g] = y2n[b * MROWS + i0 + m];      // ||p2||^2 lives in y2 (first block)
    }

    float minv[8];
    int   minj[8];
    #pragma unroll
    for (int g = 0; g < 8; ++g) { minv[g] = 3.4e38f; minj[g] = 0; }

    const float* Ybase = Y + (size_t)b * MROWS * KPAD + m16 * KPAD + 2 * h;

    auto load_tile = [&](int jt, v2f (&bb)[8], float& yn) {
        const float* yrow = Ybase + (size_t)jt * 16 * KPAD;
        #pragma unroll
        for (int s = 0; s < 8; ++s)
            bb[s] = *(const v2f*)(yrow + 4 * s);
        yn = y2n[b * MROWS + jt * 16 + m16];
    };

    auto proc_tile = [&](int jt, const v2f (&bb)[8], float yn) {
        v8f c1 = {}, c2 = {};
        #pragma unroll
        for (int s = 0; s < 8; ++s) {
            c1 = __builtin_amdgcn_wmma_f32_16x16x4_f32(false, a1[s], false, bb[s],
                                                       (short)0, c1, false, false);
            c2 = __builtin_amdgcn_wmma_f32_16x16x4_f32(false, a2[s], false, bb[s],
                                                       (short)0, c2, false, false);
        }
        int j = jt * 16 + m16;
        #pragma unroll
        for (int g = 0; g < 8; ++g) {
            float d1 = x1v[g] + yn - 2.0f * c1[g];
            float d2 = x2v[g] + yn - 2.0f * c2[g];
            float sc = fmaxf(d1, 0.0f) + fmaxf(d2, 0.0f);
            if (sc < minv[g]) { minv[g] = sc; minj[g] = j; }   // strict < keeps earliest j per lane
        }
    };

    v2f bbA[8], bbB[8];
    float ynA, ynB;
    load_tile(0, bbA, ynA);
    for (int jt = 0; jt < JT; jt += 2) {
        // warm L2 a few tiles ahead (data set is L2-resident; cheap insurance)
        __builtin_prefetch(Ybase + (size_t)(jt + 4) * 16 * KPAD, 0, 1);
        load_tile(jt + 1, bbB, ynB);           // loads overlap WMMAs below
        proc_tile(jt, bbA, ynA);
        if (jt + 2 < JT) load_tile(jt + 2, bbA, ynA);
        proc_tile(jt + 1, bbB, ynB);
    }

    // min+argmin across the 16 lanes (n dimension) of each half-wave; tie -> smaller j
    #pragma unroll
    for (int g = 0; g < 8; ++g) {
        float v = minv[g];
        int   j = minj[g];
        #pragma unroll
        for (int mask = 8; mask >= 1; mask >>= 1) {
            float vo = __shfl_xor(v, mask, 32);
            int   jo = __shfl_xor(j, mask, 32);
            if (vo < v || (vo == v && jo < j)) { v = vo; j = jo; }
        }
        if (m16 == 0) ind[b * N1 + i0 + g + 8 * h] = j;
    }
}

// ---------------- epilogue: gather best buddy, mean abs diff ------------------
__global__ void zero_out(float* out) { if (threadIdx.x == 0) out[0] = 0.0f; }

__global__ void loss_kernel(const float* __restrict__ P1, const float* __restrict__ Y,
                            const int* __restrict__ ind, float* __restrict__ out) {
    int t = blockIdx.x * blockDim.x + threadIdx.x;   // 0..16383
    int b = t >> 12, i = t & 4095;
    const float* p1r = P1 + ((size_t)b * N1 + i) * KPAD;
    const float* yr  = Y  + ((size_t)b * MROWS + ind[b * N1 + i]) * KPAD;
    float s = 0.0f;
    #pragma unroll
    for (int d = 0; d < DVAL; ++d) s += fabsf(p1r[d] - yr[d]);
    __shared__ float red[256];
    red[threadIdx.x] = s;
    __syncthreads();
    for (int off = 128; off > 0; off >>= 1) {
        if (threadIdx.x < off) red[threadIdx.x] += red[threadIdx.x + off];
        __syncthreads();
    }
    if (threadIdx.x == 0) atomicAdd(out, red[0] * INV_CNT);
}

// ---------------- launcher ----------------------------------------------------
extern "C" void kernel_launch(void* const* d_in, const int* in_sizes, int n_in,
                              void* d_out, int out_size, void* d_ws, size_t ws_size,
                              hipStream_t stream) {
    const float* x  = (const float*)d_in[0];
    const float* gt = (const float*)d_in[1];
    float* ws = (float*)d_ws;

    float* gt2 = ws;                    // 4*3*96*96   = 110592
    float* gt4 = ws + 110592;           // 4*3*48*48   =  27648
    float* P1  = ws + 138240;           // 4*4096*32   = 524288
    float* Yb  = ws + 662528;           // 4*5376*32   = 688128
    float* x1n = ws + 1350656;          // 4*4096      =  16384
    float* y2n = ws + 1367040;          // 4*5376      =  21504
    int*   ind = (int*)(ws + 1388544);  // 4*4096 ints

    // bicubic downsamples of gt
    bicubic_down<<<(12 * 96 * 96 + 255) / 256, 256, 0, stream>>>(gt, gt2, HIN, 96, 2, -1);
    bicubic_down<<<(12 * 48 * 48 + 255) / 256, 256, 0, stream>>>(gt, gt4, HIN, 48, 4, 0);

    // unfold + pack (K padded to 32) + squared norms
    unfold_pack<<<(4 * N1 + 255) / 256, 256, 0, stream>>>(x,   HIN, 64, P1, x1n, N1,    0);
    unfold_pack<<<(4 * N1 + 255) / 256, 256, 0, stream>>>(gt,  HIN, 64, Yb, y2n, MROWS, 0);
    unfold_pack<<<(4 * 1024 + 255) / 256, 256, 0, stream>>>(gt2, 96, 32, Yb, y2n, MROWS, 4096);
    unfold_pack<<<(4 * 256 + 255) / 256, 256, 0, stream>>>(gt4, 48, 16, Yb, y2n, MROWS, 5120);

    // fused WMMA pdist + argmin: one wave per (i-tile, batch)
    argmin_wmma<<<dim3(N1 / 16, B_), 32, 0, stream>>>(P1, Yb, x1n, y2n, ind);

    // scalar loss
    zero_out<<<1, 32, 0, stream>>>((float*)d_out);
    loss_kernel<<<(4 * N1) / 256, 256, 0, stream>>>(P1, Yb, ind, (float*)d_out);
}